// GCNNet_39513699123711
// MI455X (gfx1250) — compile-verified
//
#include <hip/hip_runtime.h>
#include <hip/hip_bf16.h>

// Problem constants (match reference)
constexpr int Nn = 50000;
constexpr int Ee = 1600000;
constexpr int Hh = 128;     // F_IN == H == 128
constexpr int Gg = 512;
constexpr int Cc = 5;
constexpr int Pp = 25;

typedef __attribute__((ext_vector_type(2))) float v2f;
typedef __attribute__((ext_vector_type(8))) float v8f;

// ---------------------------------------------------------------------------
// Generic zero-fill
// ---------------------------------------------------------------------------
__global__ void zero_kernel(float* __restrict__ p, int n) {
    int i = blockIdx.x * blockDim.x + threadIdx.x;
    if (i < n) p[i] = 0.0f;
}

// ---------------------------------------------------------------------------
// Degree accumulation and d^{-1/2}
// ---------------------------------------------------------------------------
__global__ void deg_kernel(const int* __restrict__ dst, float* __restrict__ deg) {
    int e = blockIdx.x * blockDim.x + threadIdx.x;
    if (e < Ee) atomicAdd(&deg[dst[e]], 1.0f);
}

__global__ void dinv_kernel(float* __restrict__ deg) {
    int i = blockIdx.x * blockDim.x + threadIdx.x;
    if (i < Nn) deg[i] = 1.0f / sqrtf(deg[i] + 1.0f);   // +1 self-loop
}

// ---------------------------------------------------------------------------
// WMMA GEMM: H[N,128] = X[N,128] @ W[128,128]  (exact fp32 via V_WMMA_F32_16X16X4_F32)
// Block = 256 threads = 8 wave32; block computes 16 rows x 128 cols.
// Wave w computes the 16x16 tile at columns [16w, 16w+16).
// A frag (16x4 f32, 2 VGPR): lanes 0-15 -> K=k+0,k+1 ; lanes 16-31 -> K=k+2,k+3
// B frag (4x16 f32, 2 VGPR): same K split, N striped across lanes 0-15 / 16-31
// D (16x16 f32, 8 VGPR): VGPR r -> M=r (lanes 0-15), M=r+8 (lanes 16-31); N = lane&15
// ---------------------------------------------------------------------------
__global__ void __launch_bounds__(256) gemm_wmma_kernel(const float* __restrict__ X,
                                                        const float* __restrict__ W,
                                                        float* __restrict__ Hout) {
    __shared__ float Wlds[Hh * Hh];           // 64 KB, well under 320 KB/WGP
    const int tid = threadIdx.x;
    // cooperative 128x128 weight stage (float4)
    for (int i = tid; i < (Hh * Hh) / 4; i += 256) {
        ((float4*)Wlds)[i] = ((const float4*)W)[i];
    }
    __syncthreads();

    const int wave = tid >> 5;
    const int lane = tid & 31;
    const int row0 = blockIdx.x * 16;         // N = 50000 = 3125 * 16, exact
    const int col0 = wave * 16;
    const int half = lane >> 4;               // 0: K+0/K+1 ; 1: K+2/K+3
    const int l16  = lane & 15;
    const int koff = half * 2;
    const int arow = row0 + l16;
    const int bcol = col0 + l16;

    const float* __restrict__ xrow = X + arow * Hh;

    v8f c = {0.f, 0.f, 0.f, 0.f, 0.f, 0.f, 0.f, 0.f};
    #pragma unroll 8
    for (int k = 0; k < Hh; k += 4) {
        v2f a, b;
        a.x = xrow[k + koff];
        a.y = xrow[k + koff + 1];
        b.x = Wlds[(k + koff) * Hh + bcol];
        b.y = Wlds[(k + koff + 1) * Hh + bcol];
        c = __builtin_amdgcn_wmma_f32_16x16x4_f32(
                /*neg_a=*/false, a, /*neg_b=*/false, b,
                /*c_mod=*/(short)0, c, /*reuse_a=*/false, /*reuse_b=*/false);
    }

    float* out = Hout + (row0 + half * 8) * Hh + col0 + l16;
    #pragma unroll
    for (int r = 0; r < 8; ++r) out[r * Hh] = c[r];
}

// ---------------------------------------------------------------------------
// Edge scatter: acc[dst] += h[src] * dinv[src]*dinv[dst]
// One wave32 per edge, 4 features/lane (float4 gather + 4 f32 atomics, L2-resident)
// ---------------------------------------------------------------------------
__global__ void __launch_bounds__(256) scatter_kernel(const float* __restrict__ h,
                                                      const float* __restrict__ dinv,
                                                      const int* __restrict__ src,
                                                      const int* __restrict__ dst,
                                                      float* __restrict__ acc) {
    const int wave = threadIdx.x >> 5;
    const int lane = threadIdx.x & 31;
    const int e = blockIdx.x * 8 + wave;      // E = 1.6M = 200000 * 8, exact
    if (e >= Ee) return;
    const int s = src[e];
    const int d = dst[e];
    const float nrm = dinv[s] * dinv[d];
    const int f0 = lane * 4;
    float4 hv = *(const float4*)(h + (long)s * Hh + f0);
    float* ap = acc + (long)d * Hh + f0;
    atomicAdd(ap + 0, hv.x * nrm);
    atomicAdd(ap + 1, hv.y * nrm);
    atomicAdd(ap + 2, hv.z * nrm);
    atomicAdd(ap + 3, hv.w * nrm);
}

// ---------------------------------------------------------------------------
// Per-node epilogue: + self-loop + bias, L2 normalize, ELU.
// One wave32 per node, 4 features/lane, shfl_xor reduction over 32 lanes.
// ---------------------------------------------------------------------------
__global__ void __launch_bounds__(256) node_post_kernel(const float* __restrict__ acc,
                                                        const float* __restrict__ h,
                                                        const float* __restrict__ dinv,
                                                        const float* __restrict__ bias,
                                                        float* __restrict__ out) {
    const int wave = threadIdx.x >> 5;
    const int lane = threadIdx.x & 31;
    const int n = blockIdx.x * 8 + wave;      // N = 50000 = 6250 * 8, exact
    if (n >= Nn) return;
    const float di = dinv[n];
    const float selfw = di * di;
    const int f0 = lane * 4;
    float4 a  = *(const float4*)(acc  + (long)n * Hh + f0);
    float4 hv = *(const float4*)(h    + (long)n * Hh + f0);
    float4 bb = *(const float4*)(bias + f0);
    float v0 = a.x + hv.x * selfw + bb.x;
    float v1 = a.y + hv.y * selfw + bb.y;
    float v2 = a.z + hv.z * selfw + bb.z;
    float v3 = a.w + hv.w * selfw + bb.w;
    float ss = v0 * v0 + v1 * v1 + v2 * v2 + v3 * v3;
    #pragma unroll
    for (int m = 16; m > 0; m >>= 1) ss += __shfl_xor(ss, m, 32);
    const float inv = 1.0f / fmaxf(sqrtf(ss), 1e-12f);
    v0 *= inv; v1 *= inv; v2 *= inv; v3 *= inv;
    // ELU (alpha = 1)
    v0 = v0 > 0.f ? v0 : (expf(v0) - 1.f);
    v1 = v1 > 0.f ? v1 : (expf(v1) - 1.f);
    v2 = v2 > 0.f ? v2 : (expf(v2) - 1.f);
    v3 = v3 > 0.f ? v3 : (expf(v3) - 1.f);
    float4 r; r.x = v0; r.y = v1; r.z = v2; r.w = v3;
    *(float4*)(out + (long)n * Hh + f0) = r;
}

// ---------------------------------------------------------------------------
// Pooling
// ---------------------------------------------------------------------------
__global__ void counts_kernel(const int* __restrict__ batch, float* __restrict__ counts) {
    int n = blockIdx.x * blockDim.x + threadIdx.x;
    if (n < Nn) atomicAdd(&counts[batch[n]], 1.0f);
}

__global__ void __launch_bounds__(256) pool_accum_kernel(const float* __restrict__ x,
                                                         const int* __restrict__ batch,
                                                         float* __restrict__ pooled) {
    const int wave = threadIdx.x >> 5;
    const int lane = threadIdx.x & 31;
    const int n = blockIdx.x * 8 + wave;
    if (n >= Nn) return;
    const int g = batch[n];
    const int f0 = lane * 4;
    float4 v = *(const float4*)(x + (long)n * Hh + f0);
    float* pp = pooled + (long)g * Hh + f0;
    atomicAdd(pp + 0, v.x);
    atomicAdd(pp + 1, v.y);
    atomicAdd(pp + 2, v.z);
    atomicAdd(pp + 3, v.w);
}

__global__ void pool_div_kernel(float* __restrict__ pooled, const float* __restrict__ counts) {
    int i = blockIdx.x * blockDim.x + threadIdx.x;
    if (i < Gg * Hh) pooled[i] /= fmaxf(counts[i / Hh], 1.0f);
}

// ---------------------------------------------------------------------------
// Prototype head: dist, sim = log((d+1)/(d+eps)), logits, softmax.
// One wave32 per graph; lane p (< 25) owns prototype p.
// ---------------------------------------------------------------------------
__global__ void __launch_bounds__(32) head_kernel(const float* __restrict__ pooled,
                                                  const float* __restrict__ proto,
                                                  const float* __restrict__ lastw,
                                                  float* __restrict__ logits,
                                                  float* __restrict__ probs,
                                                  float* __restrict__ distout) {
    __shared__ float sim_s[32];
    __shared__ float logit_s[8];
    const int g = blockIdx.x;
    const int lane = threadIdx.x;
    const float* pg = pooled + (long)g * Hh;

    float gg = 0.f;
    #pragma unroll 4
    for (int i = 0; i < Hh; ++i) { float t = pg[i]; gg += t * t; }

    float simv = 0.f;
    if (lane < Pp) {
        const float* pr = proto + (long)lane * Hh;
        float xp = 0.f, pp = 0.f;
        #pragma unroll 4
        for (int i = 0; i < Hh; ++i) { float p = pr[i]; xp += pg[i] * p; pp += p * p; }
        const float dist = gg - 2.0f * xp + pp;
        distout[(long)g * Pp + lane] = dist;
        simv = logf((dist + 1.0f) / (dist + 1e-4f));
    }
    sim_s[lane] = simv;
    __syncthreads();

    if (lane < Cc) {
        float lg = 0.f;
        #pragma unroll
        for (int p = 0; p < Pp; ++p) lg += sim_s[p] * lastw[lane * Pp + p];
        logit_s[lane] = lg;
        logits[(long)g * Cc + lane] = lg;
    }
    __syncthreads();

    if (lane < Cc) {
        float mx = logit_s[0];
        #pragma unroll
        for (int c = 1; c < Cc; ++c) mx = fmaxf(mx, logit_s[c]);
        float se = 0.f;
        #pragma unroll
        for (int c = 0; c < Cc; ++c) se += expf(logit_s[c] - mx);
        probs[(long)g * Cc + lane] = expf(logit_s[lane] - mx) / se;
    }
}

// ---------------------------------------------------------------------------
// Host-side orchestration (all launches on `stream`, graph-capture safe)
// ---------------------------------------------------------------------------
extern "C" void kernel_launch(void* const* d_in, const int* in_sizes, int n_in,
                              void* d_out, int out_size, void* d_ws, size_t ws_size,
                              hipStream_t stream) {
    const float* x     = (const float*)d_in[0];
    const int*   ei    = (const int*)d_in[1];    // [2, E]: row 0 = src, row 1 = dst
    const int*   batch = (const int*)d_in[2];
    const float* Ws[3] = { (const float*)d_in[3], (const float*)d_in[5], (const float*)d_in[7] };
    const float* bs[3] = { (const float*)d_in[4], (const float*)d_in[6], (const float*)d_in[8] };
    const float* proto = (const float*)d_in[9];
    const float* lastw = (const float*)d_in[10];

    const int* src = ei;
    const int* dst = ei + Ee;

    // Workspace layout (floats)
    float* ws     = (float*)d_ws;
    float* dinv   = ws;                          // N
    float* hbuf   = dinv + Nn;                   // N*H
    float* acc    = hbuf + (size_t)Nn * Hh;      // N*H
    float* xbuf   = acc  + (size_t)Nn * Hh;      // N*H
    float* counts = xbuf + (size_t)Nn * Hh;      // G

    // Output layout (floats, return order)
    float* logits    = (float*)d_out;                       // G*C
    float* probs     = logits + (size_t)Gg * Cc;            // G*C
    float* node_emb  = probs  + (size_t)Gg * Cc;            // N*H
    float* graph_emb = node_emb + (size_t)Nn * Hh;          // G*H
    float* distout   = graph_emb + (size_t)Gg * Hh;         // G*P

    const int T = 256;

    // Degrees -> dinv (reused by all 3 layers; graph is fixed)
    zero_kernel<<<(Nn + T - 1) / T, T, 0, stream>>>(dinv, Nn);
    deg_kernel<<<(Ee + T - 1) / T, T, 0, stream>>>(dst, dinv);
    dinv_kernel<<<(Nn + T - 1) / T, T, 0, stream>>>(dinv);

    const float* xin = x;
    for (int layer = 0; layer < 3; ++layer) {
        float* xout = (layer == 2) ? node_emb : xbuf;
        gemm_wmma_kernel<<<Nn / 16, T, 0, stream>>>(xin, Ws[layer], hbuf);
        zero_kernel<<<((Nn * Hh) + T - 1) / T, T, 0, stream>>>(acc, Nn * Hh);
        scatter_kernel<<<Ee / 8, T, 0, stream>>>(hbuf, dinv, src, dst, acc);
        node_post_kernel<<<Nn / 8, T, 0, stream>>>(acc, hbuf, dinv, bs[layer], xout);
        xin = xout;
    }

    // global_mean_pool
    zero_kernel<<<(Gg + T - 1) / T, T, 0, stream>>>(counts, Gg);
    zero_kernel<<<((Gg * Hh) + T - 1) / T, T, 0, stream>>>(graph_emb, Gg * Hh);
    counts_kernel<<<(Nn + T - 1) / T, T, 0, stream>>>(batch, counts);
    pool_accum_kernel<<<Nn / 8, T, 0, stream>>>(node_emb, batch, graph_emb);
    pool_div_kernel<<<((Gg * Hh) + T - 1) / T, T, 0, stream>>>(graph_emb, counts);

    // prototype distances + head + softmax
    head_kernel<<<Gg, 32, 0, stream>>>(graph_emb, proto, lastw, logits, probs, distout);
}